// GNNAgent_48490180772147
// MI455X (gfx1250) — compile-verified
//
#include <hip/hip_runtime.h>
#include <math.h>

#define N_NODES 100000
#define N_EDGES 1600000
#define D_IN    64
#define D_HID   64
#define D_ACT   16

typedef __attribute__((ext_vector_type(2))) float v2f;
typedef __attribute__((ext_vector_type(8))) float v8f;

// LDS stride for the 64x64 weight tile: 80 floats/row.
// bank(k,n) = (80k + n) % 64 = (16k + n) % 64 -> the two wave halves
// (k offset by 2 -> +32 banks) read disjoint bank sets: conflict-free.
#define WSTRIDE 80

// ---------------- degree / dinv ----------------

__global__ void k_deg_init(float* __restrict__ deg, int n) {
  int i = blockIdx.x * blockDim.x + threadIdx.x;
  if (i < n) deg[i] = 1.0f;                    // self-loop
}

__global__ void k_deg_accum(const long long* __restrict__ dst,
                            float* __restrict__ deg, int ne) {
  int e = blockIdx.x * blockDim.x + threadIdx.x;
  if (e < ne) atomicAdd(&deg[(int)dst[e]], 1.0f);
}

__global__ void k_dinv(const float* __restrict__ deg,
                       float* __restrict__ dinv, int n) {
  int i = blockIdx.x * blockDim.x + threadIdx.x;
  if (i < n) dinv[i] = rsqrtf(deg[i]);
}

// ---------------- WMMA GEMM: [N,64] x [64,64] ----------------
// Each wave: 16-row tile, full 64 output cols (4 n-tiles), K=64.
// RELU template: A-load applies relu(A + abias) (fused prev-layer epilogue).
// Stores H = A@W and ACC = H * dinv[row]^2 (ACC may alias A in-place:
// each wave reads only its own 16 A-rows, and all loads feed the WMMAs
// before the epilogue stores).

template <bool RELU>
__global__ __launch_bounds__(128) void k_gemm64(
    const float* __restrict__ A, const float* __restrict__ abias,
    const float* __restrict__ W, const float* __restrict__ dinv,
    float* __restrict__ H, float* __restrict__ ACC, int nrows)
{
  __shared__ float Wlds[64 * WSTRIDE];
  __shared__ float blds[64];

  const int tid = threadIdx.x;

  // stage W (64x64) into LDS, padded rows; float4 per trip, 8 trips/thread
  for (int i4 = tid; i4 < 1024; i4 += 128) {
    const int lin = i4 * 4;
    const int k = lin >> 6, n = lin & 63;
    const float4 w = *(const float4*)(W + lin);
    *(float4*)(&Wlds[k * WSTRIDE + n]) = w;
  }
  if (RELU) {
    if (tid < 64) blds[tid] = abias[tid];
  }
  __syncthreads();

  const int lane = tid & 31;
  const int wave = tid >> 5;
  const int tile = blockIdx.x * 4 + wave;
  const int rowBase = tile * 16;
  if (rowBase >= nrows) return;              // wave-uniform: EXEC stays all-1s

  const int m  = lane & 15;                  // row within tile (A) / col (B,C)
  const int kh = lane >> 4;                  // k-half selector
  const int row = rowBase + m;

  v8f c0 = {}, c1 = {}, c2 = {}, c3 = {};

  for (int kk = 0; kk < 64; kk += 4) {
    const int k0 = kk + 2 * kh;
    // A fragment: a[j] = A[row][k0 + j]
    const float* ap = A + (size_t)row * 64 + k0;
    v2f a;
    a.x = ap[0];
    a.y = ap[1];
    if (RELU) {
      a.x = fmaxf(a.x + blds[k0],     0.0f);
      a.y = fmaxf(a.y + blds[k0 + 1], 0.0f);
    }
    // B fragments from LDS: b[j] = W[k0 + j][nBase + m]
    const float* wp = &Wlds[k0 * WSTRIDE + m];
    v2f b0 = { wp[0],  wp[WSTRIDE] };
    v2f b1 = { wp[16], wp[WSTRIDE + 16] };
    v2f b2 = { wp[32], wp[WSTRIDE + 32] };
    v2f b3 = { wp[48], wp[WSTRIDE + 48] };
    c0 = __builtin_amdgcn_wmma_f32_16x16x4_f32(false, a, false, b0, (short)0, c0, false, false);
    c1 = __builtin_amdgcn_wmma_f32_16x16x4_f32(false, a, false, b1, (short)0, c1, false, false);
    c2 = __builtin_amdgcn_wmma_f32_16x16x4_f32(false, a, false, b2, (short)0, c2, false, false);
    c3 = __builtin_amdgcn_wmma_f32_16x16x4_f32(false, a, false, b3, (short)0, c3, false, false);
  }

  // C layout: VGPR v -> row = rowBase + v + 8*kh, col = nBase + m
#pragma unroll
  for (int v = 0; v < 8; ++v) {
    const int r = rowBase + v + 8 * kh;
    const float d = dinv[r];
    const float d2 = d * d;
    float* hp = H   + (size_t)r * 64 + m;
    float* op = ACC + (size_t)r * 64 + m;
    const float h0 = c0[v], h1 = c1[v], h2 = c2[v], h3 = c3[v];
    hp[0]  = h0;  op[0]  = h0 * d2;
    hp[16] = h1;  op[16] = h1 * d2;
    hp[32] = h2;  op[32] = h2 * d2;
    hp[48] = h3;  op[48] = h3 * d2;
  }
}

// ---------------- edge scatter: ACC[dst] += coef * H[src] ----------------
// 16 threads per edge, float4 per thread (coalesced 256B gather per edge).

__global__ __launch_bounds__(256) void k_scatter(
    const float* __restrict__ H, const float* __restrict__ dinv,
    const long long* __restrict__ src, const long long* __restrict__ dst,
    float* __restrict__ ACC, int ne)
{
  const long long gid = (long long)blockIdx.x * blockDim.x + threadIdx.x;
  const int e = (int)(gid >> 4);
  const int q = (int)(gid & 15);
  if (e >= ne) return;
  const int s = (int)src[e];
  const int d = (int)dst[e];
  const float coef = dinv[s] * dinv[d];
  const float4 hv = *(const float4*)(H + (size_t)s * 64 + q * 4);
  float* op = ACC + (size_t)d * 64 + q * 4;
  atomicAdd(op + 0, hv.x * coef);
  atomicAdd(op + 1, hv.y * coef);
  atomicAdd(op + 2, hv.z * coef);
  atomicAdd(op + 3, hv.w * coef);
}

// ---------------- FC head: out = relu(ACC + b2) @ Wfc + bfc ----------------
// Wfc is 64x16; natural LDS stride 16 gives the two wave halves a +32 bank
// separation -> conflict-free without padding.

__global__ __launch_bounds__(128) void k_fc(
    const float* __restrict__ A, const float* __restrict__ abias,
    const float* __restrict__ W,   // [64,16]
    const float* __restrict__ ob,  // [16]
    float* __restrict__ OUT, int nrows)
{
  __shared__ float Wlds[64 * 16];
  __shared__ float blds[64];

  const int tid = threadIdx.x;
  for (int i4 = tid; i4 < 256; i4 += 128) {
    *(float4*)(&Wlds[i4 * 4]) = *(const float4*)(W + i4 * 4);
  }
  if (tid < 64) blds[tid] = abias[tid];
  __syncthreads();

  const int lane = tid & 31;
  const int wave = tid >> 5;
  const int tile = blockIdx.x * 4 + wave;
  const int rowBase = tile * 16;
  if (rowBase >= nrows) return;

  const int m  = lane & 15;
  const int kh = lane >> 4;
  const int row = rowBase + m;

  v8f c = {};
  for (int kk = 0; kk < 64; kk += 4) {
    const int k0 = kk + 2 * kh;
    const float* ap = A + (size_t)row * 64 + k0;
    v2f a;
    a.x = fmaxf(ap[0] + blds[k0],     0.0f);
    a.y = fmaxf(ap[1] + blds[k0 + 1], 0.0f);
    const float* wp = &Wlds[k0 * 16 + m];
    v2f b = { wp[0], wp[16] };
    c = __builtin_amdgcn_wmma_f32_16x16x4_f32(false, a, false, b, (short)0, c, false, false);
  }

  const float bb = ob[m];
#pragma unroll
  for (int v = 0; v < 8; ++v) {
    const int r = rowBase + v + 8 * kh;
    OUT[(size_t)r * 16 + m] = c[v] + bb;
  }
}

// ---------------- launch ----------------

extern "C" void kernel_launch(void* const* d_in, const int* in_sizes, int n_in,
                              void* d_out, int out_size, void* d_ws, size_t ws_size,
                              hipStream_t stream) {
  (void)in_sizes; (void)n_in; (void)out_size; (void)ws_size;

  const float*     x   = (const float*)d_in[0];
  const long long* ei  = (const long long*)d_in[1];   // [2, N_EDGES] int64
  const float*     W1  = (const float*)d_in[2];
  const float*     b1  = (const float*)d_in[3];
  const float*     W2  = (const float*)d_in[4];
  const float*     b2  = (const float*)d_in[5];
  const float*     Wfc = (const float*)d_in[6];
  const float*     bfc = (const float*)d_in[7];
  float* out = (float*)d_out;

  const long long* src = ei;
  const long long* dst = ei + N_EDGES;

  // workspace layout
  char* ws = (char*)d_ws;
  float* deg  = (float*)(ws);                                   // 400 KB
  float* dinv = (float*)(ws + (512u << 10));                    // 400 KB
  float* H    = (float*)(ws + (1u << 20));                      // 25.6 MB
  float* ACC  = (float*)(ws + (1u << 20) + (size_t)N_NODES * 64 * 4); // 25.6 MB

  const int tiles  = (N_NODES + 15) / 16;        // 6250
  const int gblk   = (tiles + 3) / 4;            // 4 waves / block
  const int nblk   = (N_NODES + 255) / 256;
  const int eblk   = (N_EDGES + 255) / 256;
  const long long sthreads = (long long)N_EDGES * 16;
  const int sblk   = (int)((sthreads + 255) / 256);

  // degrees
  k_deg_init <<<nblk, 256, 0, stream>>>(deg, N_NODES);
  k_deg_accum<<<eblk, 256, 0, stream>>>(dst, deg, N_EDGES);
  k_dinv     <<<nblk, 256, 0, stream>>>(deg, dinv, N_NODES);

  // layer 1: h1 = x @ W1 ; acc1 = h1*dinv^2 ; acc1 += scatter
  k_gemm64<false><<<gblk, 128, 0, stream>>>(x, nullptr, W1, dinv, H, ACC, N_NODES);
  k_scatter      <<<sblk, 256, 0, stream>>>(H, dinv, src, dst, ACC, N_EDGES);

  // layer 2: A = relu(acc1 + b1), in-place reuse of H/ACC
  k_gemm64<true> <<<gblk, 128, 0, stream>>>(ACC, b1, W2, dinv, H, ACC, N_NODES);
  k_scatter      <<<sblk, 256, 0, stream>>>(H, dinv, src, dst, ACC, N_EDGES);

  // head: out = relu(acc2 + b2) @ Wfc + bfc
  k_fc<<<gblk, 128, 0, stream>>>(ACC, b2, Wfc, bfc, out, N_NODES);
}